// Probe_12790412607932
// MI455X (gfx1250) — compile-verified
//
#include <hip/hip_runtime.h>
#include <hip/hip_bf16.h>

#define Bn 8192
#define Tn 2048
#define Cn 8
#define ROW (Tn * Cn)            // 16384 floats per batch row (64 KB)
#define TILE_ELTS 4096           // 16 KB tile
#define NTILES (ROW / TILE_ELTS) // 4
#define THREADS 256

typedef unsigned int u32x4 __attribute__((ext_vector_type(4)));
typedef int          i32x8 __attribute__((ext_vector_type(8)));
typedef int          i32x4 __attribute__((ext_vector_type(4)));
typedef float        v2f   __attribute__((ext_vector_type(2)));
typedef float        v8f   __attribute__((ext_vector_type(8)));

#ifndef __has_builtin
#define __has_builtin(x) 0
#endif

#if __has_builtin(__builtin_amdgcn_tensor_load_to_lds)
#define HAVE_TDM 1
#else
#define HAVE_TDM 0
#endif

template <int N>
__device__ __forceinline__ void wait_tensorcnt() {
#if __has_builtin(__builtin_amdgcn_s_wait_tensorcnt)
    __builtin_amdgcn_s_wait_tensorcnt(N);   // N is an ICE via template parameter
#else
    asm volatile("s_wait_tensorcnt %0" :: "i"(N) : "memory");
#endif
}

// Issue one TDM bulk copy: n8 8-byte elements from gsrc (global) to LDS byte
// offset lds_off. Wave-scalar op: caller ensures only one wave executes it.
__device__ __forceinline__ void tdm_load(const float* gsrc, unsigned lds_off, unsigned n8) {
#if HAVE_TDM
    unsigned long long ga = (unsigned long long)gsrc;
    u32x4 g0;
    g0.x = 1u;                                   // count=1 (valid descriptor), no gather
    g0.y = lds_off;                              // lds_addr (bytes)
    g0.z = (unsigned)ga;                         // global_addr[31:0]
    g0.w = (unsigned)(ga >> 32) | (2u << 30);    // global_addr[56:32] | type=2
    i32x8 g1;
    g1[0] = (3 << 16);                           // data_size=3 (8 bytes), no multicast
    g1[1] = (int)((n8 & 0xFFFFu) << 16);         // tensor_dim0[15:0]
    g1[2] = (int)((n8 >> 16) | (1u << 16));      // tensor_dim0[31:16] | tensor_dim1=1
    g1[3] = (int)((n8 & 0xFFFFu) << 16);         // tile_dim0 = n8
    g1[4] = 1;                                   // tile_dim1 = 1
    g1[5] = (int)n8;                             // tensor_dim0_stride
    g1[6] = 0;
    g1[7] = 0;
    i32x4 g2 = {0, 0, 0, 0};                     // 2D only: groups 2/3 unused
    i32x4 g3 = {0, 0, 0, 0};
    i32x8 g4 = {0, 0, 0, 0, 0, 0, 0, 0};         // 6-arg toolchain form: extra group
    __builtin_amdgcn_tensor_load_to_lds(g0, g1, g2, g3, g4, 0);
#else
    (void)gsrc; (void)lds_off; (void)n8;         // unused: kernel takes direct-load path
#endif
}

__device__ __forceinline__ void top3_update(float f, float& a0, float& a1, float& a2) {
    if (f > a0)      { a2 = a1; a1 = a0; a0 = f; }
    else if (f > a1) { a2 = a1; a1 = f; }
    else if (f > a2) { a2 = f; }
}

// Phase 1: one block per batch row. TDM streams 16 KB tiles into LDS
// (double buffered, TENSORcnt-synced); threads maintain per-channel top-3;
// LDS reduction produces feats[b][c*4 + {top0,top1,top2,mean}].
__global__ __launch_bounds__(THREADS) void topk_feats_kernel(const float* __restrict__ x,
                                                             float* __restrict__ feats) {
    __shared__ __align__(16) float buf[2][TILE_ELTS];   // 32 KB
    __shared__ float red[Cn * 128 * 3];                 // 12 KB

    const int b    = blockIdx.x;
    const int L    = threadIdx.x;
    const int wave = L >> 5;
    const float* row = x + (unsigned long long)b * ROW;

    // Low 32 bits of a generic pointer to LDS == workgroup-relative LDS byte addr
    const unsigned lds0 = (unsigned)(unsigned long long)(void*)&buf[0][0];
    const unsigned lds1 = (unsigned)(unsigned long long)(void*)&buf[1][0];

    float t0[4], t1[4], t2[4];
#pragma unroll
    for (int j = 0; j < 4; ++j) { t0[j] = t1[j] = t2[j] = -3.402823466e38f; }

#if HAVE_TDM
    if (wave == 0) {
        tdm_load(row, lds0, TILE_ELTS / 2);             // tile 0 -> buf[0]
    }
    for (int i = 0; i < NTILES; ++i) {
        if (wave == 0) {
            if (i + 1 < NTILES) {
                tdm_load(row + (i + 1) * TILE_ELTS, ((i + 1) & 1) ? lds1 : lds0,
                         TILE_ELTS / 2);                // prefetch tile i+1
                wait_tensorcnt<1>();                    // oldest (tile i) complete
            } else {
                wait_tensorcnt<0>();
            }
        }
        __syncthreads();                                // tile i visible to all waves
        const float4* src = (const float4*)&buf[i & 1][0];
#pragma unroll
        for (int q = 0; q < TILE_ELTS / 4 / THREADS; ++q) {
            float4 v = src[q * THREADS + L];
            top3_update(v.x, t0[0], t1[0], t2[0]);
            top3_update(v.y, t0[1], t1[1], t2[1]);
            top3_update(v.z, t0[2], t1[2], t2[2]);
            top3_update(v.w, t0[3], t1[3], t2[3]);
        }
        __syncthreads();                                // done reading before overwrite
    }
#else
    // No TDM builtin: stream straight from global (same lane->channel mapping).
    for (int i = 0; i < NTILES; ++i) {
        const float4* src = (const float4*)(row + i * TILE_ELTS);
#pragma unroll
        for (int q = 0; q < TILE_ELTS / 4 / THREADS; ++q) {
            float4 v = src[q * THREADS + L];
            top3_update(v.x, t0[0], t1[0], t2[0]);
            top3_update(v.y, t0[1], t1[1], t2[1]);
            top3_update(v.z, t0[2], t1[2], t2[2]);
            top3_update(v.w, t0[3], t1[3], t2[3]);
        }
    }
#endif

    // Each thread's float4 covers channels 4p..4p+3 where p = L&1
    // (element index 4*(q*256+L), modulo 8 => 4*(L&1)).
    const int p    = L & 1;
    const int slot = L >> 1;                            // 0..127 per channel group
#pragma unroll
    for (int j = 0; j < 4; ++j) {
        float* dst = &red[((4 * p + j) * 128 + slot) * 3];
        dst[0] = t0[j]; dst[1] = t1[j]; dst[2] = t2[j];
    }
    __syncthreads();

    if (L < Cn) {
        float a0 = -3.402823466e38f, a1 = a0, a2 = a0;
        const float* src = &red[L * 128 * 3];
        for (int s = 0; s < 128 * 3; ++s) top3_update(src[s], a0, a1, a2);
        float m = (a0 + a1 + a2) * (1.0f / 3.0f);
        float* fo = feats + (unsigned long long)b * 32 + L * 4;
        fo[0] = a0; fo[1] = a1; fo[2] = a2; fo[3] = m;
    }
}

// Phase 2: out[16 rows per wave] = feats(16x32) . W(32) + bias, all fp32 via
// V_WMMA_F32_16X16X4_F32, accumulating 8 chunks of K=4. W is broadcast across
// all 16 B-columns, so every column of D holds the GEMV result.
__global__ __launch_bounds__(32) void gemv_wmma_kernel(const float* __restrict__ feats,
                                                       const float* __restrict__ W,
                                                       const float* __restrict__ bias,
                                                       float* __restrict__ out) {
    const int lane    = threadIdx.x;        // 0..31, EXEC all ones (no divergence)
    const int rowBase = blockIdx.x * 16;
    const int m       = lane & 15;          // A-matrix row for this lane
    const int half    = lane >> 4;          // 0: K={0,1}, 1: K={2,3} within chunk

    const float* arow = feats + (unsigned long long)(rowBase + m) * 32 + 2 * half;
    v8f acc = {};
#pragma unroll
    for (int ch = 0; ch < 8; ++ch) {
        v2f a, bm;
        a.x  = arow[ch * 4 + 0];
        a.y  = arow[ch * 4 + 1];
        bm.x = W[ch * 4 + 2 * half + 0];    // B[k][n] = W[ch*4+k] for all n
        bm.y = W[ch * 4 + 2 * half + 1];
        acc = __builtin_amdgcn_wmma_f32_16x16x4_f32(false, a, false, bm,
                                                    (short)0, acc, false, false);
    }
    const float bs = bias[0];
    // D col 0: lane 0 holds rows 0..7 in acc[0..7]; lane 16 holds rows 8..15.
    if (m == 0) {
        float* o = out + rowBase + 8 * half;
#pragma unroll
        for (int r = 0; r < 8; ++r) o[r] = acc[r] + bs;
    }
}

extern "C" void kernel_launch(void* const* d_in, const int* in_sizes, int n_in,
                              void* d_out, int out_size, void* d_ws, size_t ws_size,
                              hipStream_t stream) {
    const float* x    = (const float*)d_in[0];   // [8192, 2048, 8] f32
    const float* W    = (const float*)d_in[1];   // [1, 32] f32
    const float* bias = (const float*)d_in[2];   // [1] f32
    float* out   = (float*)d_out;                // [8192] f32
    float* feats = (float*)d_ws;                 // [8192, 32] f32 scratch (1 MB)

    topk_feats_kernel<<<Bn, THREADS, 0, stream>>>(x, feats);
    gemv_wmma_kernel<<<Bn / 16, 32, 0, stream>>>(feats, W, bias, out);
}